// Griddle_66279935311944
// MI455X (gfx1250) — compile-verified
//
#include <hip/hip_runtime.h>
#include <math.h>

// ---------------------------------------------------------------------------
// Reference:  B=524288 rows, T=24.
//   mix[24][8] = states(24x3) @ E(3x8)           (tiny, once)
//   out[b]     = prod_t mix[t][ob[b][t]-1]       (memory-bound stream, 52 MB)
// MI455X: 23.3 TB/s HBM -> ~2.2us floor; everything else must be free.
// ---------------------------------------------------------------------------

typedef float v2f __attribute__((ext_vector_type(2)));
typedef float v8f __attribute__((ext_vector_type(8)));

#define T_STEPS 24

__device__ inline void softmax_to(const float* g, int n, float* dst, int npad) {
    float mx = g[0];
    for (int i = 1; i < n; ++i) mx = fmaxf(mx, g[i]);
    float sum = 0.0f;
    for (int i = 0; i < n; ++i) { float e = expf(g[i] - mx); dst[i] = e; sum += e; }
    float inv = 1.0f / sum;
    for (int i = 0; i < n; ++i) dst[i] *= inv;
    for (int i = n; i < npad; ++i) dst[i] = 0.0f;
}

// ---------------------------------------------------------------------------
// Kernel A: one wave32. Serial scan on lane 0 into LDS, then
// mix(24x8) = states(24x3) @ E(3x8) via two V_WMMA_F32_16X16X4_F32
// (K padded 3->4 with zeros; call 1 covers rows 0..15, call 2 rows 16..23).
// ---------------------------------------------------------------------------
__global__ __launch_bounds__(32)
void Griddle_mix_precompute(const float* __restrict__ grid4,
                            const float* __restrict__ grid6,
                            const float* __restrict__ grid8,
                            const float* __restrict__ init_state,
                            const float* __restrict__ state_M,
                            float* __restrict__ mix_out /* 24*8 floats */) {
    __shared__ float sE[3][8];         // E rows (softmaxed grids, zero-padded)
    __shared__ float sS[T_STEPS][3];   // states emitted by the scan

    const int lane = threadIdx.x;
    if (lane == 0) {
        softmax_to(grid4, 4, &sE[0][0], 8);
        softmax_to(grid6, 6, &sE[1][0], 8);
        softmax_to(grid8, 8, &sE[2][0], 8);

        float s[3];
        softmax_to(init_state, 3, s, 3);
        for (int t = 0; t < T_STEPS; ++t) {
            sS[t][0] = s[0]; sS[t][1] = s[1]; sS[t][2] = s[2];
            float v[3];
            for (int j = 0; j < 3; ++j)
                v[j] = s[0] * state_M[0 * 3 + j] +
                       s[1] * state_M[1 * 3 + j] +
                       s[2] * state_M[2 * 3 + j];
            softmax_to(v, 3, s, 3);
        }
    }
    __syncthreads();

    const int  mrow = lane & 15;
    const int  col  = lane & 15;
    const bool hi   = lane >= 16;   // lanes 16..31 carry K=2 (VGPR0) / K=3 (VGPR1)

#if __has_builtin(__builtin_amdgcn_wmma_f32_16x16x4_f32)
    // A fragment, 32-bit 16x4 layout: VGPR0 = K0 (lanes 0-15) | K2 (lanes 16-31),
    //                                 VGPR1 = K1 (lanes 0-15) | K3 (lanes 16-31).
    v2f a1, a2, b;
    a1.x = hi ? sS[mrow][2] : sS[mrow][0];
    a1.y = hi ? 0.0f        : sS[mrow][1];

    float a2x = 0.0f, a2y = 0.0f;      // call 2: states rows 16..23 in M=0..7
    if (mrow < 8) {
        a2x = hi ? sS[16 + mrow][2] : sS[16 + mrow][0];
        a2y = hi ? 0.0f             : sS[16 + mrow][1];
    }
    a2.x = a2x; a2.y = a2y;

    // B fragment, 4x16 mirror layout: cols 8..15 zero-padded.
    float b0 = (col < 8) ? sE[0][col] : 0.0f;
    float b1 = (col < 8) ? sE[1][col] : 0.0f;
    float b2 = (col < 8) ? sE[2][col] : 0.0f;
    b.x = hi ? b2 : b0;
    b.y = hi ? 0.0f : b1;

    v8f c = {};
    v8f d1 = __builtin_amdgcn_wmma_f32_16x16x4_f32(false, a1, false, b,
                                                   (short)0, c, false, false);
    v8f d2 = __builtin_amdgcn_wmma_f32_16x16x4_f32(false, a2, false, b,
                                                   (short)0, c, false, false);

    // D layout: VGPR v holds (M=v, N=lane) for lanes 0-15, (M=v+8, N=lane-16) hi.
    if (col < 8) {
#pragma unroll
        for (int v = 0; v < 8; ++v) {
            mix_out[(v + (hi ? 8 : 0)) * 8 + col] = d1[v];
            if (!hi) mix_out[(16 + v) * 8 + col] = d2[v];
        }
    }
#else
    // Scalar fallback (keeps file compiling if the builtin signature differs).
    for (int t = lane; t < T_STEPS; t += 32)
        for (int j = 0; j < 8; ++j)
            mix_out[t * 8 + j] = sS[t][0] * sE[0][j] +
                                 sS[t][1] * sE[1][j] +
                                 sS[t][2] * sE[2][j];
#endif
}

// ---------------------------------------------------------------------------
// Kernel B: one thread per row. 6x int4 (96 B/row) coalesced loads; the
// 192-float mix table lives in 6 VGPRs per lane (flat = r*32 + lane) and is
// gathered with ds_bpermute_b32: for step t the source lane is
// (t%4)*8 + (ob-1), which never crosses a 32-lane register boundary.
// ---------------------------------------------------------------------------
__global__ __launch_bounds__(256)
void Griddle_prob_prod(const int* __restrict__ obs,
                       const float* __restrict__ mixws,
                       float* __restrict__ out, int B) {
    const int row  = blockIdx.x * 256 + threadIdx.x;
    const int lane = threadIdx.x & 31;

    // Table registers: mreg[r] on lane l = mix_flat[r*32 + l]  (24*8 = 192 = 6*32)
    float mreg[6];
#pragma unroll
    for (int r = 0; r < 6; ++r) mreg[r] = mixws[r * 32 + lane];

    // Clamp instead of early-return so every lane stays EXEC-on through the
    // bpermutes (disabled source lanes would read back zero).
    const long long rr = (row < B) ? (long long)row : (long long)(B - 1);
    const int4* o4 = reinterpret_cast<const int4*>(obs + rr * T_STEPS);

    int ob[T_STEPS];
#pragma unroll
    for (int r = 0; r < 6; ++r) {
        int4 q = o4[r];
        ob[4 * r + 0] = q.x; ob[4 * r + 1] = q.y;
        ob[4 * r + 2] = q.z; ob[4 * r + 3] = q.w;
    }

    float p = 1.0f;
#pragma unroll
    for (int t = 0; t < T_STEPS; ++t) {
        // byte index = ((t%4)*8 + (ob-1)) * 4, always in [0,128)
        int byteidx = ((t & 3) << 5) + (ob[t] << 2) - 4;
        int bits = __builtin_amdgcn_ds_bpermute(byteidx,
                                                __float_as_int(mreg[t >> 2]));
        p *= __int_as_float(bits);
    }

    if (row < B) out[row] = p;
}

// ---------------------------------------------------------------------------
extern "C" void kernel_launch(void* const* d_in, const int* in_sizes, int n_in,
                              void* d_out, int out_size, void* d_ws, size_t ws_size,
                              hipStream_t stream) {
    (void)n_in; (void)out_size; (void)ws_size;
    const int*   obs  = (const int*)  d_in[0];
    const float* g4   = (const float*)d_in[1];
    const float* g6   = (const float*)d_in[2];
    const float* g8   = (const float*)d_in[3];
    const float* s0   = (const float*)d_in[4];
    const float* M    = (const float*)d_in[5];
    float*       outp = (float*)d_out;
    float*       mixw = (float*)d_ws;   // 24*8 floats = 768 bytes of scratch

    const int B = in_sizes[0] / T_STEPS;

    hipLaunchKernelGGL(Griddle_mix_precompute, dim3(1), dim3(32), 0, stream,
                       g4, g6, g8, s0, M, mixw);

    const int blocks = (B + 255) / 256;
    hipLaunchKernelGGL(Griddle_prob_prod, dim3(blocks), dim3(256), 0, stream,
                       obs, mixw, outp, B);
}